// GlobalImageLevelLoss_28484223107533
// MI455X (gfx1250) — compile-verified
//
#include <hip/hip_runtime.h>
#include <hip/hip_bf16.h>

#define B_    128
#define P_    16
#define N_    32
#define D_    1024
#define MARGIN_ 0.2f
#define ROWSTRIDE 68   // 64 + 4 pad: conflict-free LDS banks, 16B-aligned rows

typedef float v2f __attribute__((ext_vector_type(2)));
typedef float v4f __attribute__((ext_vector_type(4)));
typedef float v8f __attribute__((ext_vector_type(8)));

// One block = (batch b, group g of 4 positives). 8 waves; wave w owns the
// 16-row tile (g*8+w) of negatives[b] (512 rows total per b).
__global__ __launch_bounds__(256)
void GlobalImageLevelLoss_main(const float* __restrict__ img,
                               const float* __restrict__ pos,
                               const float* __restrict__ neg,
                               float* __restrict__ partial) {
    __shared__ float imgLds[D_];                   //  4 KB
    __shared__ float negLds[8][16 * ROWSTRIDE];    // 34 KB
    __shared__ float posPart[256];                 //  1 KB
    __shared__ float tileMax[8];
    __shared__ float hingeV[4];

    const int tid  = threadIdx.x;
    const int b    = blockIdx.y;   // 0..127
    const int g    = blockIdx.x;   // 0..3
    const int w    = tid >> 5;     // wave 0..7
    const int lane = tid & 31;

    // ---- stage img_feats[b] into LDS: 256 threads x float4, fully coalesced
    {
        const v4f* src = (const v4f*)(img + (size_t)b * D_);
        ((v4f*)imgLds)[tid] = src[tid];
    }
    __syncthreads();

    // ---- positive similarities (4 p per block, 64 threads each), VALU path
    {
        const int pl  = tid >> 6;       // 0..3
        const int l64 = tid & 63;
        const float* row = pos + ((size_t)b * P_ + (g * 4 + pl)) * D_;
        float s = 0.f;
        #pragma unroll 4
        for (int d = l64; d < D_; d += 64) s += row[d] * imgLds[d];
        posPart[tid] = s;               // read after the barrier below
    }

    // ---- negatives via WMMA
    const int   tileIdx = g * 8 + w;                 // 0..31 per b
    const int   rbase   = tileIdx * 16;              // global row in [0,512)
    const float* negBase = neg + ((size_t)b * (P_ * N_) + rbase) * D_;
    float* myLds = negLds[w];

    const int rowS = lane >> 4;          // staging: 2 rows per pass
    const int f4   = lane & 15;          // float4 slot within a row-panel
    const int arow = lane & 15;          // WMMA A: M = lane % 16
    const int koff = (lane >> 4) << 1;   // WMMA A/B: K base = 2*(lane>=16)

    v8f acc = {0.f, 0.f, 0.f, 0.f, 0.f, 0.f, 0.f, 0.f};

    for (int kb = 0; kb < D_; kb += 64) {
        // stage one 16x64 fp32 panel: 8 passes x (2 rows x 16 float4), coalesced b128
        #pragma unroll
        for (int pp = 0; pp < 8; ++pp) {
            const int row = pp * 2 + rowS;
            v4f v = *(const v4f*)(negBase + (size_t)row * D_ + kb + f4 * 4);
            *(v4f*)(myLds + row * ROWSTRIDE + f4 * 4) = v;
        }
        if (kb + 64 < D_)   // hint next panel into cache (global_prefetch_b8)
            __builtin_prefetch(negBase + (size_t)rowS * D_ + (kb + 64) + f4 * 4, 0, 1);
        // per-wave LDS ordering + compiler fence (LDS is in-order per wave)
        asm volatile("s_wait_dscnt 0" ::: "memory");

        // 16 chained K=4 fp32 WMMAs; B = img chunk replicated over all 16 cols
        #pragma unroll
        for (int c = 0; c < 16; ++c) {
            v2f a  = *(const v2f*)(myLds + arow * ROWSTRIDE + c * 4 + koff);
            v2f bv = *(const v2f*)(imgLds + kb + c * 4 + koff);
            acc = __builtin_amdgcn_wmma_f32_16x16x4_f32(
                      false, a, false, bv, (short)0, acc, false, false);
        }
        asm volatile("s_wait_dscnt 0" ::: "memory");
    }

    // every column of D equals the row dot-product -> max over 8 regs + wave max
    float m = acc[0];
    #pragma unroll
    for (int i = 1; i < 8; ++i) m = fmaxf(m, acc[i]);
    #pragma unroll
    for (int i = 16; i >= 1; i >>= 1) m = fmaxf(m, __shfl_xor(m, i, 32));
    if (lane == 0) tileMax[w] = m;      // max over this tile's 16 negatives
    __syncthreads();

    // combine: p_local = tid (0..3) owns tiles 2p, 2p+1 (n 0-15, 16-31)
    if (tid < 4) {
        float ps = 0.f;
        #pragma unroll
        for (int i = 0; i < 64; ++i) ps += posPart[tid * 64 + i];
        const float mn = fmaxf(tileMax[2 * tid], tileMax[2 * tid + 1]);
        hingeV[tid] = fmaxf(0.f, MARGIN_ + mn - ps);
    }
    __syncthreads();
    if (tid == 0)
        partial[b * 4 + g] = hingeV[0] + hingeV[1] + hingeV[2] + hingeV[3];
}

// Deterministic fixed-order tree reduction of the 512 block partials.
__global__ __launch_bounds__(256)
void GlobalImageLevelLoss_reduce(const float* __restrict__ partial,
                                 float* __restrict__ out) {
    __shared__ float s[256];
    const int tid = threadIdx.x;
    s[tid] = partial[tid] + partial[tid + 256];
    __syncthreads();
    for (int off = 128; off > 0; off >>= 1) {
        if (tid < off) s[tid] += s[tid + off];
        __syncthreads();
    }
    if (tid == 0) out[0] = s[0];
}

extern "C" void kernel_launch(void* const* d_in, const int* in_sizes, int n_in,
                              void* d_out, int out_size, void* d_ws, size_t ws_size,
                              hipStream_t stream) {
    const float* img = (const float*)d_in[0];   // (128, 1024)
    const float* pos = (const float*)d_in[1];   // (128, 16, 1024)
    const float* neg = (const float*)d_in[2];   // (128, 16, 32, 1024)
    float* out     = (float*)d_out;             // scalar loss
    float* partial = (float*)d_ws;              // 512 floats of scratch

    dim3 grid(4, B_);
    GlobalImageLevelLoss_main<<<grid, 256, 0, stream>>>(img, pos, neg, partial);
    GlobalImageLevelLoss_reduce<<<1, 256, 0, stream>>>(partial, out);
}